// GCNLayer_8985071583979
// MI455X (gfx1250) — compile-verified
//
#include <hip/hip_runtime.h>
#include <hip/hip_bf16.h>

#define IN_F   128
#define OUT_F  128
#define LN_EPS 1e-5f

typedef float v2f __attribute__((ext_vector_type(2)));
typedef float v8f __attribute__((ext_vector_type(8)));

// ---------------------------------------------------------------------------
// Edge scatter: ah[dst] += h[src] (128 f32 per edge), deg[dst] += 1.
// 32 lanes per edge, each lane handles a float4 slice -> 4 HW f32 atomics.
// h (51MB) and ah (51MB) both fit in the 192MB L2, so this runs at L2 speed.
// ---------------------------------------------------------------------------
__global__ void __launch_bounds__(256)
gcn_edge_scatter(const float* __restrict__ h,
                 const int*   __restrict__ src,
                 const int*   __restrict__ dst,
                 float* ah, float* deg, int n_edges)
{
    const int e = blockIdx.x * 8 + (threadIdx.x >> 5);
    if (e >= n_edges) return;
    const int lane = threadIdx.x & 31;

    const int s = src[e];
    const int d = dst[e];

    const float4 v = ((const float4*)(h + (size_t)s * IN_F))[lane];
    float* dp = ah + (size_t)d * IN_F + lane * 4;
    unsafeAtomicAdd(dp + 0, v.x);
    unsafeAtomicAdd(dp + 1, v.y);
    unsafeAtomicAdd(dp + 2, v.z);
    unsafeAtomicAdd(dp + 3, v.w);
    if (lane == 0) unsafeAtomicAdd(&deg[d], 1.0f);
}

// ---------------------------------------------------------------------------
// Fused: x = (ah * 1/deg) @ W + bias ; LayerNorm(feature dim) ; ReLU.
// 256 threads = 8 waves; each wave computes a 16x128 row block with
// V_WMMA_F32_16X16X4_F32 (8 N-tiles, K-loop of 32 steps).
//
// W is staged in LDS k-pair interleaved + tile-blocked:
//   sW[t*2048 + k2*32 + c16*2 + j]  holds  W[(2*k2+j)*128 + t*16 + c16]
// so each B operand is ONE contiguous ds_load_b64 into an even VGPR pair
// (no re-pairing movs), and the 32-dword k2 stride puts lanes 0-15 and
// 16-31 in disjoint bank halves -> conflict-free across all 64 LDS banks.
//
// In-place: ah == out (each wave reads all K of its rows before storing).
// ---------------------------------------------------------------------------
__global__ void __launch_bounds__(256)
gcn_gemm_ln(const float* ah,
            const float* __restrict__ deg,
            const float* __restrict__ W,
            const float* __restrict__ bias,
            const float* __restrict__ gamma,
            const float* __restrict__ beta,
            float* out, int n_nodes)
{
    __shared__ float sW[IN_F * OUT_F];   // 64 KB of the 320 KB/WGP LDS
    for (int i = threadIdx.x; i < IN_F * OUT_F; i += 256) {
        const int t   = i >> 11;        // output-col tile   (0..7)
        const int rem = i & 2047;
        const int k2  = rem >> 5;       // k-pair index      (0..63)
        const int w   = rem & 31;
        const int c16 = w >> 1;         // col within tile   (0..15)
        const int j   = w & 1;          // k parity
        sW[i] = W[(size_t)(2 * k2 + j) * OUT_F + (t * 16 + c16)];
    }
    __syncthreads();

    const int wave = threadIdx.x >> 5;
    const int lane = threadIdx.x & 31;
    const int half = lane >> 4;          // 0: lanes 0-15, 1: lanes 16-31
    const int l16  = lane & 15;

    const int m0 = (blockIdx.x * 8 + wave) * 16;  // first row of this wave
    if (m0 >= n_nodes) return;                    // whole-wave exit

    // A-operand row for this lane (lanes 16-31 reuse rows with K offset +2)
    const int arow = m0 + l16;
    const float dv  = deg[arow];
    const float nrm = dv > 0.0f ? 1.0f / dv : 0.0f;
    const float* aptr = ah + (size_t)arow * IN_F + 2 * half;

    v8f acc[8];
#pragma unroll
    for (int t = 0; t < 8; ++t)
        acc[t] = (v8f){0.f, 0.f, 0.f, 0.f, 0.f, 0.f, 0.f, 0.f};

    // K loop: 128 in steps of 4 (f32 WMMA shape 16x16x4)
    for (int kk = 0; kk < IN_F; kk += 4) {
        v2f a = *(const v2f*)(aptr + kk);   // A[row][kk+2*half .. +1]
        a.x *= nrm;
        a.y *= nrm;
        // k-pair for this lane: lanes 0-15 -> kk/2, lanes 16-31 -> kk/2+1
        const float* bbase = sW + ((kk >> 1) + half) * 32 + l16 * 2;
#pragma unroll
        for (int t = 0; t < 8; ++t) {
            const v2f b = *(const v2f*)(bbase + t * 2048);  // ds_load_b64
            acc[t] = __builtin_amdgcn_wmma_f32_16x16x4_f32(
                false, a, false, b, (short)0, acc[t], false, false);
        }
    }

    // Per-column epilogue constants (col = t*16 + l16)
    float biascol[8], gcol[8], btcol[8];
#pragma unroll
    for (int t = 0; t < 8; ++t) {
        const int c = t * 16 + l16;
        biascol[t] = bias[c];
        gcol[t]    = gamma[c];
        btcol[t]   = beta[c];
    }
#pragma unroll
    for (int t = 0; t < 8; ++t)
#pragma unroll
        for (int r = 0; r < 8; ++r)
            acc[t][r] += biascol[t];

    // LayerNorm + ReLU. VGPR r of lanes 0-15 holds row m0+r (cols = lane),
    // lanes 16-31 hold row m0+8+r. xor-shuffles over masks 1..8 reduce the
    // 16 per-lane partials of each row without crossing lane halves.
#pragma unroll
    for (int r = 0; r < 8; ++r) {
        float s = 0.f, q = 0.f;
#pragma unroll
        for (int t = 0; t < 8; ++t) {
            const float v = acc[t][r];
            s += v;
            q += v * v;
        }
#pragma unroll
        for (int m = 1; m < 16; m <<= 1) {
            s += __shfl_xor(s, m, 32);
            q += __shfl_xor(q, m, 32);
        }
        const float mu   = s * (1.0f / OUT_F);
        const float var  = q * (1.0f / OUT_F) - mu * mu;
        const float rstd = rsqrtf(var + LN_EPS);

        float* orow = out + (size_t)(m0 + r + 8 * half) * OUT_F + l16;
#pragma unroll
        for (int t = 0; t < 8; ++t) {
            float y = (acc[t][r] - mu) * rstd * gcol[t] + btcol[t];
            y = y > 0.0f ? y : 0.0f;
            orow[t * 16] = y;
        }
    }
}

// ---------------------------------------------------------------------------
extern "C" void kernel_launch(void* const* d_in, const int* in_sizes, int n_in,
                              void* d_out, int out_size, void* d_ws, size_t ws_size,
                              hipStream_t stream)
{
    const float* h     = (const float*)d_in[0];
    const float* W     = (const float*)d_in[1];
    const float* bias  = (const float*)d_in[2];
    const float* gamma = (const float*)d_in[3];
    const float* beta  = (const float*)d_in[4];
    const int*   src   = (const int*)d_in[5];
    const int*   dst   = (const int*)d_in[6];

    const int n_nodes = in_sizes[0] / IN_F;
    const int n_edges = in_sizes[5];

    float* ah  = (float*)d_out;   // aggregation accumulator lives in d_out
    float* deg = (float*)d_ws;    // n_nodes floats of scratch

    hipMemsetAsync(ah,  0, (size_t)n_nodes * IN_F * sizeof(float), stream);
    hipMemsetAsync(deg, 0, (size_t)n_nodes * sizeof(float), stream);

    gcn_edge_scatter<<<(n_edges + 7) / 8, 256, 0, stream>>>(
        h, src, dst, ah, deg, n_edges);

    const int mblocks = (n_nodes + 15) / 16;        // 16-row blocks
    gcn_gemm_ln<<<(mblocks + 7) / 8, 256, 0, stream>>>(
        ah, deg, W, bias, gamma, beta, (float*)d_out, n_nodes);
}